// QueryBasedAggregation_42391327212214
// MI455X (gfx1250) — compile-verified
//
#include <hip/hip_runtime.h>
#include <stdint.h>

// Problem shape (from reference)
#define D_MODEL   256
#define N_PATCH   196
#define TILE_ELEMS (D_MODEL * N_PATCH)   // 50176 f32 = 196 KB (fits 320 KB LDS)
#define BLOCK     256                    // 8 wave32 waves
#define CHUNK_ROWS 49                    // 196 / 4 chunks
#define NCHUNK    4

typedef __attribute__((ext_vector_type(2))) float        v2f;
typedef __attribute__((ext_vector_type(8))) float        v8f;
typedef __attribute__((ext_vector_type(4))) unsigned int v4u;
typedef __attribute__((ext_vector_type(8))) int          v8i;
typedef __attribute__((ext_vector_type(4))) int          v4i;

#if defined(__gfx1250__) && __has_builtin(__builtin_amdgcn_tensor_load_to_lds)
#define HAS_TDM 1
#else
#define HAS_TDM 0
#endif

#if defined(__gfx1250__) && __has_builtin(__builtin_amdgcn_wmma_f32_16x16x4_f32)
#define HAS_WMMA 1
#else
#define HAS_WMMA 0
#endif

#if HAS_TDM
// Build a 2D-tile TDM descriptor (D#) per CDNA5 ISA and
// issue TENSOR_LOAD_TO_LDS. Copies `rows` x 256 f32, contiguous in both global
// and LDS. data_size=4B, no pad, no iterate, not in a cluster (wg_mask=0).
__device__ __forceinline__ void tdm_load_rows(uint32_t lds_byte, uint64_t gaddr,
                                              uint32_t rows) {
  v4u g0;
  g0[0] = 1u;                                   // count=1 valid descriptor
  g0[1] = lds_byte;                             // lds_addr (wave-relative bytes)
  g0[2] = (uint32_t)(gaddr & 0xFFFFFFFFu);      // global_addr[31:0]
  g0[3] = (uint32_t)((gaddr >> 32) & 0x01FFFFFFu) | (2u << 30); // addr[56:32] | type=2

  v8i g1;
  g1[0] = (int)(2u << 16);                      // wg_mask=0, data_size=2 (4B)
  g1[1] = (int)((uint32_t)D_MODEL << 16);       // tensor_dim0[15:0] @ bits63:48
  g1[2] = (int)((rows & 0xFFFFu) << 16);        // tensor_dim0 hi=0 | tensor_dim1 lo
  g1[3] = (int)((uint32_t)D_MODEL << 16);       // tensor_dim1 hi=0 | tile_dim0=256
  g1[4] = (int)(rows & 0xFFFFu);                // tile_dim1=rows, tile_dim2=0
  g1[5] = (int)D_MODEL;                         // tensor_dim0_stride = 256 elems
  g1[6] = 0;                                    // stride0 hi | stride1 lo (unused)
  g1[7] = 0;

  v4i g2; g2[0] = 1; g2[1] = 1; g2[2] = 0; g2[3] = 0;   // dim2=dim3=1 (unused)
  v4i g3; g3[0] = 0; g3[1] = (int)(1u << 16); g3[2] = 0; g3[3] = 0; // dim4=1

#if __clang_major__ >= 23
  v8i g4 = {0, 0, 0, 0, 0, 0, 0, 0};
  __builtin_amdgcn_tensor_load_to_lds(g0, g1, g2, g3, g4, 0);
#else
  __builtin_amdgcn_tensor_load_to_lds(g0, g1, g2, g3, 0);
#endif
}
#endif // HAS_TDM

__device__ __forceinline__ float wave_sum32(float v) {
  v += __shfl_xor(v, 16, 32);
  v += __shfl_xor(v, 8, 32);
  v += __shfl_xor(v, 4, 32);
  v += __shfl_xor(v, 2, 32);
  v += __shfl_xor(v, 1, 32);
  return v;
}
__device__ __forceinline__ float wave_max32(float v) {
  v = fmaxf(v, __shfl_xor(v, 16, 32));
  v = fmaxf(v, __shfl_xor(v, 8, 32));
  v = fmaxf(v, __shfl_xor(v, 4, 32));
  v = fmaxf(v, __shfl_xor(v, 2, 32));
  v = fmaxf(v, __shfl_xor(v, 1, 32));
  return v;
}

__global__ __launch_bounds__(BLOCK) void qagg_kernel(
    const float* __restrict__ h_p, const float* __restrict__ query,
    const float* __restrict__ pe, float* __restrict__ out) {
  // Dynamic LDS: tile(196*256 f32) | s_part(2*196 -> 400) | s_beta(208)
  extern __shared__ float smem[];
  float* tile   = smem;
  float* s_part = smem + TILE_ELEMS;
  float* s_beta = s_part + 400;

  const int tid  = threadIdx.x;
  const int lane = tid & 31;
  const int wave = tid >> 5;      // 0..7, uniform per wave
  const int bn   = blockIdx.x;    // (b,n) pair
  const float* hp_base = h_p + (size_t)bn * TILE_ELEMS;

  // Column-major lane layout for the streaming pass: 64 lanes per row,
  // each lane owns a fixed float4 column -> q chunk loaded once per thread.
  const int col = tid & 63;       // float4 column 0..63
  const int r   = tid >> 6;       // row-in-group 0..3 (uniform per wave)
  const float4 q4 = ((const float4*)query)[col];

#if HAS_TDM
  // Wave 0 issues 4 chained TDM copies (HBM -> LDS). TDM ops from one wave
  // complete in order, so s_wait_tensorcnt(N-1-c) gates chunk c.
  if (__builtin_amdgcn_readfirstlane(wave) == 0) {
    const uint32_t lds0 = (uint32_t)(uintptr_t)tile; // flat[31:0] == LDS offset
    const uint64_t g0   = (uint64_t)(uintptr_t)hp_base;
    const uint32_t step = CHUNK_ROWS * D_MODEL * 4;
    tdm_load_rows(lds0 + 0 * step, g0 + (uint64_t)0 * step, CHUNK_ROWS);
    tdm_load_rows(lds0 + 1 * step, g0 + (uint64_t)1 * step, CHUNK_ROWS);
    tdm_load_rows(lds0 + 2 * step, g0 + (uint64_t)2 * step, CHUNK_ROWS);
    tdm_load_rows(lds0 + 3 * step, g0 + (uint64_t)3 * step, CHUNK_ROWS);
  }
#endif

  float4* tile4 = (float4*)tile;
  const float4* pe4 = (const float4*)pe;

  // Per chunk: wait for DMA, then fused (h + pe) rewrite + row-dot with q.
  for (int c = 0; c < NCHUNK; ++c) {
#if HAS_TDM
    if (__builtin_amdgcn_readfirstlane(wave) == 0) {
      if (c == 0)      __builtin_amdgcn_s_wait_tensorcnt(3);
      else if (c == 1) __builtin_amdgcn_s_wait_tensorcnt(2);
      else if (c == 2) __builtin_amdgcn_s_wait_tensorcnt(1);
      else             __builtin_amdgcn_s_wait_tensorcnt(0);
    }
#else
    // Fallback: cooperative vectorized copy of chunk c.
    {
      const int base = c * CHUNK_ROWS * (D_MODEL / 4);
      const float4* hp4 = (const float4*)hp_base;
      for (int t = tid; t < CHUNK_ROWS * (D_MODEL / 4); t += BLOCK)
        tile4[base + t] = hp4[base + t];
    }
#endif
    __syncthreads();

    for (int it = 0; it < 13; ++it) {
      const int pl = it * 4 + r;          // uniform per wave
      if (pl < CHUNK_ROWS) {              // wave-uniform predicate
        const int p   = c * CHUNK_ROWS + pl;
        const int idx = p * (D_MODEL / 4) + col;
        float4 h  = tile4[idx];
        float4 pv = pe4[idx];
        float4 v  = make_float4(h.x + pv.x, h.y + pv.y, h.z + pv.z, h.w + pv.w);
        tile4[idx] = v;                   // tile now holds h_pe
        float dp = v.x * q4.x + v.y * q4.y + v.z * q4.z + v.w * q4.w;
        dp = wave_sum32(dp);
        if (lane == 0) s_part[2 * p + (wave & 1)] = dp; // deterministic 2-way split
      }
    }
  }
  __syncthreads();

  // Softmax over 196 scores (wave 0 only; trivial work).
  if (wave == 0) {
    float mx = -3.0e38f;
    for (int p = lane; p < N_PATCH; p += 32) {
      float sv = (s_part[2 * p] + s_part[2 * p + 1]) * 0.0625f; // 1/sqrt(256)
      s_beta[p] = sv;
      mx = fmaxf(mx, sv);
    }
    mx = wave_max32(mx);
    float sum = 0.0f;
    for (int p = lane; p < N_PATCH; p += 32) {
      float e = expf(s_beta[p] - mx);
      s_beta[p] = e;
      sum += e;
    }
    sum = wave_sum32(sum);
    const float inv = 1.0f / sum;
    for (int p = lane; p < N_PATCH; p += 32) s_beta[p] *= inv;
  }
  __syncthreads();

  // Phase 2: out[d] = sum_p beta[p] * h_pe[p,d], reduced with WMMA.
  // A (16x4 f32, documented layout): A[m,k] = beta[p0+k]*tile[p0+k, d0+m]
  //   lane<16 : M=lane,  VGPR0->K=0, VGPR1->K=1
  //   lane>=16: M=lane-16, VGPR0->K=2, VGPR1->K=3
  // B = all-ones (every element 1.0f -> immune to B lane-layout details)
  // D[m,n] = sum_k A[m,k]; accumulate over 49 K-chunks in C.
#if HAS_WMMA
  {
    const int hi = lane >> 4;   // 0: K=0,1  1: K=2,3
    const int mc = lane & 15;   // M / output column within 16-wide d chunk
    v2f ones; ones[0] = 1.0f; ones[1] = 1.0f;
    for (int j = 0; j < 2; ++j) {
      const int d0 = (wave * 2 + j) * 16;   // 8 waves * 2 = 16 chunks = 256 d
      v8f acc = {0.f, 0.f, 0.f, 0.f, 0.f, 0.f, 0.f, 0.f};
      for (int k = 0; k < 49; ++k) {
        const int pr = 4 * k + 2 * hi;
        const float b0 = s_beta[pr];
        const float b1 = s_beta[pr + 1];
        const float t0 = tile[pr * D_MODEL + d0 + mc];
        const float t1 = tile[(pr + 1) * D_MODEL + d0 + mc];
        v2f a; a[0] = b0 * t0; a[1] = b1 * t1;
        acc = __builtin_amdgcn_wmma_f32_16x16x4_f32(
            false, a, false, ones, (short)0, acc, false, false);
      }
      // C/D layout: lane<16 VGPRj = D[j, lane]; lane>=16 VGPRj = D[8+j, lane-16].
      // All n-columns of D are identical; lanes 0 and 16 emit 8 results each.
      if (mc == 0) {
        float* op = out + (size_t)bn * D_MODEL + d0 + hi * 8;
        float4* o4 = (float4*)op;
        o4[0] = make_float4(acc[0], acc[1], acc[2], acc[3]);
        o4[1] = make_float4(acc[4], acc[5], acc[6], acc[7]);
      }
    }
  }
#else
  {
    // Fallback: lane-per-d accumulation (conflict-free LDS reads).
    float acc = 0.0f;
    const int d = tid;
    for (int p = 0; p < N_PATCH; ++p) acc += s_beta[p] * tile[p * D_MODEL + d];
    out[(size_t)bn * D_MODEL + d] = acc;
  }
#endif
}

extern "C" void kernel_launch(void* const* d_in, const int* in_sizes, int n_in,
                              void* d_out, int out_size, void* d_ws, size_t ws_size,
                              hipStream_t stream) {
  (void)in_sizes; (void)n_in; (void)d_ws; (void)ws_size;
  const float* h_p   = (const float*)d_in[0];
  const float* query = (const float*)d_in[1];
  const float* pe    = (const float*)d_in[2];
  float* out = (float*)d_out;

  const int nbn = out_size / D_MODEL;  // 32*64 = 2048 (b,n) pairs
  const size_t shmem = (size_t)(TILE_ELEMS + 400 + 208) * sizeof(float); // ~198.4 KB

  // >64KB dynamic LDS: opt in (CDNA5 WGP has 320 KB). Deterministic, not a
  // stream op, safe under graph capture.
  (void)hipFuncSetAttribute((const void*)qagg_kernel,
                            hipFuncAttributeMaxDynamicSharedMemorySize,
                            (int)shmem);

  qagg_kernel<<<nbn, BLOCK, shmem, stream>>>(h_p, query, pe, out);
}